// TreeLSTM_61349312856636
// MI455X (gfx1250) — compile-verified
//
#include <hip/hip_runtime.h>
#include <math.h>

// TreeLSTM, 4-ary, depth 8. HID = input_dim = 256, 4H = 1024.
// One kernel per level (8 stream-ordered launches). Each 512-thread WG
// (16 waves) computes a 32-node M-tile of
//   iou = x @ wx + ch_h @ wh + bx + bh
// with V_WMMA_F32_16X16X4_F32 (exact fp32). Each wave owns a 16-wide hcol
// slice replicated over the 4 gates (4 col-tiles) x 2 row-tiles, so every
// B fragment is reused by two WMMAs -> halves L2 weight traffic vs M=16.

#define HID    256
#define FOURH  1024
#define NPAD   260          // LDS row stride (floats): +4 kills bank conflicts
#define MT     32           // nodes per workgroup (2 row-tiles of 16)
#define NNODES 21845

typedef float v2f __attribute__((ext_vector_type(2)));
typedef float v8f __attribute__((ext_vector_type(8)));

__device__ __forceinline__ float sigf(float x) {
  return 1.0f / (1.0f + expf(-x));
}

__global__ __launch_bounds__(512)
void treelstm_level_kernel(const float* __restrict__ emb,
                           const float* __restrict__ wx,
                           const float* __restrict__ bx,
                           const float* __restrict__ wh,
                           const float* __restrict__ bh,
                           float* __restrict__ h_all,
                           float* __restrict__ c_all,
                           float* __restrict__ root_out,
                           int level_start, int child_start,
                           int n, int is_leaf, int is_root)
{
  __shared__ float s_a[MT][NPAD];     // staged A tile: x first, then sum(child h)

  const int tid  = threadIdx.x;
  const int lane = tid & 31;
  const int w    = tid >> 5;          // wave 0..15 -> hcol slice [16w, 16w+16)
  const int m    = lane & 15;
  const int hi   = lane >> 4;
  const int node_base = blockIdx.x * MT;

  // ---- Stage embedding tile (zero-pad rows past n) ----
  for (int e = tid; e < MT * HID; e += 512) {
    const int row = e >> 8;           // HID == 256
    const int kk  = e & (HID - 1);
    const int node = node_base + row;
    s_a[row][kk] = (node < n) ? emb[(size_t)(level_start + node) * HID + kk] : 0.0f;
  }
  __syncthreads();

  // col-tile t = gate (i,o,u,f); acc[r][t] is row-tile r (rows 16r..16r+15)
  int col[4];
  v8f acc[2][4];
#pragma unroll
  for (int t = 0; t < 4; ++t) {
    col[t] = (t << 8) + (w << 4) + m;
    const float bias = bx[col[t]] + bh[col[t]];
    v8f z = {bias, bias, bias, bias, bias, bias, bias, bias};
    acc[0][t] = z;
    acc[1][t] = z;
  }

  // ---- x @ wx : K steps of 4; each B fragment feeds both row-tiles ----
  for (int k0 = 0; k0 < HID; k0 += 4) {
    const int kk = k0 + (hi << 1);
    v2f a0, a1;
    a0.x = s_a[m][kk];        a0.y = s_a[m][kk + 1];
    a1.x = s_a[16 + m][kk];   a1.y = s_a[16 + m][kk + 1];
    const float* b0 = wx + (size_t)kk * FOURH;
#pragma unroll
    for (int t = 0; t < 4; ++t) {
      v2f b;
      b.x = b0[col[t]];
      b.y = b0[FOURH + col[t]];
      acc[0][t] = __builtin_amdgcn_wmma_f32_16x16x4_f32(
          false, a0, false, b, (short)0, acc[0][t], false, false);
      acc[1][t] = __builtin_amdgcn_wmma_f32_16x16x4_f32(
          false, a1, false, b, (short)0, acc[1][t], false, false);
    }
  }

  // ---- ch_h @ wh (internal levels; block-uniform branch) ----
  if (!is_leaf) {
    __syncthreads();                  // all waves done reading x tile
    for (int e = tid; e < MT * HID; e += 512) {
      const int row = e >> 8;
      const int kk  = e & (HID - 1);
      const int node = node_base + row;
      float hv = 0.0f;
      if (node < n) {
        const size_t cb = (size_t)(child_start + (node << 2)) * HID + kk;
        hv = h_all[cb] + h_all[cb + HID] + h_all[cb + 2 * HID] + h_all[cb + 3 * HID];
      }
      s_a[row][kk] = hv;
    }
    __syncthreads();

    for (int k0 = 0; k0 < HID; k0 += 4) {
      const int kk = k0 + (hi << 1);
      v2f a0, a1;
      a0.x = s_a[m][kk];        a0.y = s_a[m][kk + 1];
      a1.x = s_a[16 + m][kk];   a1.y = s_a[16 + m][kk + 1];
      const float* b0 = wh + (size_t)kk * FOURH;
#pragma unroll
      for (int t = 0; t < 4; ++t) {
        v2f b;
        b.x = b0[col[t]];
        b.y = b0[FOURH + col[t]];
        acc[0][t] = __builtin_amdgcn_wmma_f32_16x16x4_f32(
            false, a0, false, b, (short)0, acc[0][t], false, false);
        acc[1][t] = __builtin_amdgcn_wmma_f32_16x16x4_f32(
            false, a1, false, b, (short)0, acc[1][t], false, false);
      }
    }
  }

  // ---- Gates in-register: t = 0:i, 1:o, 2:u, 3:f ----
#pragma unroll
  for (int r = 0; r < 2; ++r) {
#pragma unroll
    for (int v = 0; v < 8; ++v) {
      const int row  = (r << 4) + (hi << 3) + v;   // D layout: vgpr v -> row v+8*hi
      const int node = node_base + row;
      if (node < n) {
        const int hcol = (w << 4) + m;
        const float ig = sigf (acc[r][0][v]);
        const float og = sigf (acc[r][1][v]);
        const float ug = tanhf(acc[r][2][v]);
        const float fg = sigf (acc[r][3][v]);
        float cc = ig * ug;
        if (!is_leaf) {
          const size_t cb = (size_t)(child_start + (node << 2)) * HID + hcol;
          const float cs = c_all[cb] + c_all[cb + HID] +
                           c_all[cb + 2 * HID] + c_all[cb + 3 * HID];
          cc += fg * cs;
        }
        const float hh = og * tanhf(cc);
        const size_t o = (size_t)(level_start + node) * HID + hcol;
        h_all[o] = hh;
        c_all[o] = cc;
        if (is_root && node == 0) {
          root_out[hcol]       = hh;   // h first
          root_out[HID + hcol] = cc;   // then c
        }
      }
    }
  }
}

extern "C" void kernel_launch(void* const* d_in, const int* in_sizes, int n_in,
                              void* d_out, int out_size, void* d_ws, size_t ws_size,
                              hipStream_t stream) {
  (void)in_sizes; (void)n_in; (void)out_size; (void)ws_size;
  const float* emb = (const float*)d_in[0];  // [21845, 256]
  const float* wx  = (const float*)d_in[1];  // [256, 1024]
  const float* bx  = (const float*)d_in[2];  // [1024]
  const float* wh  = (const float*)d_in[3];  // [256, 1024]
  const float* bh  = (const float*)d_in[4];  // [1024]
  float* out   = (float*)d_out;              // h[256] ++ c[256]
  float* h_all = (float*)d_ws;               // [21845, 256]
  float* c_all = h_all + (size_t)NNODES * HID;

  // level_starts[l] = (4^l - 1) / 3
  int starts[9];
  {
    int s = 0, pw = 1;
    for (int l = 0; l <= 8; ++l) { starts[l] = s; s += pw; pw <<= 2; }
  }

  for (int l = 7; l >= 0; --l) {
    const int n    = 1 << (2 * l);
    const int grid = (n + MT - 1) / MT;
    treelstm_level_kernel<<<grid, 512, 0, stream>>>(
        emb, wx, bx, wh, bh, h_all, c_all, out,
        starts[l], starts[l + 1], n,
        (l == 7) ? 1 : 0, (l == 0) ? 1 : 0);
  }
}